// CompactPiecewiseLinearEmbeddings0d_73504070304189
// MI455X (gfx1250) — compile-verified
//
#include <hip/hip_runtime.h>

// CompactPiecewiseLinearEmbeddings — CDNA5 (gfx1250) WMMA f32 16x16x4 version.
// out[n, f*8+d] = sum_b clamp(h[n,f,b]) * W[f,b,d] + bias[f,d]

typedef __attribute__((ext_vector_type(2))) float v2f;
typedef __attribute__((ext_vector_type(8))) float v8f;

#define N_ROWS 16384
#define NF     256
#define NB     48
#define ND     8
#define FTILE  8      // features per block (one per wave; 8 waves)
#define NTILES 8      // 16-row n-tiles per block
#define OUTPAD 68     // padded LDS row stride (bank-conflict avoidance)

__global__ __launch_bounds__(256) void cple_wmma_kernel(
    const float* __restrict__ x,      // (N, F)
    const float* __restrict__ edges,  // (F, B)
    const float* __restrict__ width,  // (F, B)
    const float* __restrict__ W,      // (F, B, D)
    const float* __restrict__ bias,   // (F, D)
    float* __restrict__ out)          // (N, F*D)
{
    __shared__ float sE[FTILE * NB];
    __shared__ float sWd[FTILE * NB];
    __shared__ float sW[FTILE * NB * ND];
    __shared__ float sB[FTILE * ND];
    __shared__ float sOut[16 * OUTPAD];

    const int tid   = threadIdx.x;
    const int wave  = tid >> 5;       // 0..7 -> feature within tile
    const int lane  = tid & 31;
    const int half  = lane >> 4;      // 0: K/M low half, 1: high half
    const int lrow  = lane & 15;      // A: row M; B/C: column N (= d)
    const int fBase = blockIdx.y * FTILE;
    const int f     = fBase + wave;

    // ---- stage tables for 8 consecutive features into LDS (all contiguous) ----
    {
        const float4* gW = (const float4*)(W + (size_t)fBase * NB * ND);
        float4* s = (float4*)sW;
        #pragma unroll
        for (int i = tid; i < FTILE * NB * ND / 4; i += 256) s[i] = gW[i];
        if (tid < FTILE * NB / 4) {
            ((float4*)sE)[tid]  = ((const float4*)(edges + fBase * NB))[tid];
            ((float4*)sWd)[tid] = ((const float4*)(width + fBase * NB))[tid];
        }
        if (tid < FTILE * ND / 4)
            ((float4*)sB)[tid] = ((const float4*)(bias + fBase * ND))[tid];
    }
    __syncthreads();

    // ---- hoist per-lane operands (edges, 1/width, W column) into registers ----
    // A-matrix 16x4 f32 layout: lanes 0-15 hold K={0,1} (vgpr0,1), lanes 16-31 K={2,3}.
    // B-matrix 4x16 mirrors it: this lane's W rows are the SAME bins it computes h for.
    float eA[12][2], iA[12][2], wv[12][2];
    #pragma unroll
    for (int k = 0; k < 12; ++k) {
        #pragma unroll
        for (int j = 0; j < 2; ++j) {
            const int bin = 4 * k + 2 * half + j;
            eA[k][j] = sE[wave * NB + bin];
            iA[k][j] = 1.0f / sWd[wave * NB + bin];
            wv[k][j] = (lrow < ND) ? sW[(wave * NB + bin) * ND + lrow] : 0.0f;
        }
    }
    const float bv = (lrow < ND) ? sB[wave * ND + lrow] : 0.0f;

    // ---- n-tile loop: 12x chained V_WMMA_F32_16X16X4_F32 per 16x8 output tile ----
    for (int t = 0; t < NTILES; ++t) {
        const int nBase = (blockIdx.x * NTILES + t) * 16;
        const float xv = x[(size_t)(nBase + lrow) * NF + f];

        v8f c;
        #pragma unroll
        for (int r = 0; r < 8; ++r) c[r] = bv;   // bias folded into accumulator

        #pragma unroll
        for (int k = 0; k < 12; ++k) {
            float h[2];
            #pragma unroll
            for (int j = 0; j < 2; ++j) {
                float hv = (xv - eA[k][j]) * iA[k][j];
                // bin 0: min(h,1) only; bin 47: max(h,0) only; middle: clamp[0,1]
                float lo = 0.0f, hi = 1.0f;
                if (k == 0 && j == 0)  lo = (half == 0) ? -__builtin_inff() : 0.0f;
                if (k == 11 && j == 1) hi = (half == 1) ?  __builtin_inff() : 1.0f;
                h[j] = fminf(fmaxf(hv, lo), hi);
            }
            v2f a;  a.x = h[0];     a.y = h[1];
            v2f bm; bm.x = wv[k][0]; bm.y = wv[k][1];
            c = __builtin_amdgcn_wmma_f32_16x16x4_f32(
                    false, a, false, bm, (short)0, c, false, false);
        }

        // ---- transpose 8 waves' 16x8 tiles through LDS for coalesced stores ----
        __syncthreads();   // previous tile's sOut fully consumed
        if (lrow < ND) {
            #pragma unroll
            for (int r = 0; r < 8; ++r)   // C layout: vgpr r -> row r + 8*half, col lrow
                sOut[(r + 8 * half) * OUTPAD + wave * ND + lrow] = c[r];
        }
        __syncthreads();

        // 16 rows x 64 consecutive floats (8 features x 8 dims) -> coalesced
        #pragma unroll
        for (int i = 0; i < 4; ++i) {
            const int idx = tid + i * 256;
            const int row = idx >> 6, col = idx & 63;
            out[(size_t)(nBase + row) * (NF * ND) + blockIdx.y * 64 + col] =
                sOut[row * OUTPAD + col];
        }
    }
}

extern "C" void kernel_launch(void* const* d_in, const int* in_sizes, int n_in,
                              void* d_out, int out_size, void* d_ws, size_t ws_size,
                              hipStream_t stream) {
    const float* x     = (const float*)d_in[0];
    const float* edges = (const float*)d_in[1];
    const float* width = (const float*)d_in[2];
    const float* W     = (const float*)d_in[3];
    const float* b     = (const float*)d_in[4];
    float* out = (float*)d_out;

    dim3 grid(N_ROWS / (16 * NTILES), NF / FTILE);   // (128, 32) = 4096 blocks
    dim3 block(256);                                  // 8 wave32 waves
    cple_wmma_kernel<<<grid, block, 0, stream>>>(x, edges, width, W, b, out);
}